// ScaledDotProductAttention_82523501625734
// MI455X (gfx1250) — compile-verified
//
#include <hip/hip_runtime.h>

typedef __attribute__((ext_vector_type(2))) float v2f;
typedef __attribute__((ext_vector_type(8))) float v8f;

#define B_      2
#define H_      16
#define L_      2048
#define D_      128
#define SCALE_  0.08838834764831845f   /* 1/sqrt(128) */
#define MASKF_  -1e9f
#define NKT     (L_ / 16)              /* 128 key tiles / query tiles */
#define THREADS 256
#define NWAVES  8

__global__ __launch_bounds__(THREADS)
void sdpa_fused_f32wmma(const float* __restrict__ Q,
                        const float* __restrict__ K,
                        const float* __restrict__ V,
                        float* __restrict__ ctx,
                        float* __restrict__ attn)
{
    extern __shared__ float smem[];
    float* Ssm = smem;             // [16][2048] score / prob row block (128 KB)
    float* Qsm = smem + 16 * L_;   // [16][128]  query tile (8 KB)

    const int tid  = threadIdx.x;
    const int lane = tid & 31;
    const int wave = tid >> 5;
    const int qt   = blockIdx.x & (NKT - 1);   // query tile index in [0,128)
    const int bh   = blockIdx.x >> 7;          // b*H + h in [0,32)
    const int qrow0 = qt * 16;

    const float* Qb = Q + (size_t)bh * L_ * D_;
    const float* Kb = K + (size_t)bh * L_ * D_;
    const float* Vb = V + (size_t)bh * L_ * D_;

    // ---- stage Q tile (16x128) into LDS ----
    for (int i = tid; i < 16 * D_; i += THREADS) {
        int r = i >> 7, c = i & 127;
        Qsm[i] = Qb[(size_t)(qrow0 + r) * D_ + c];
    }
    __syncthreads();

    // WMMA 16x16x4 f32 lane decomposition (ISA 7.12.2):
    //   A: lane holds M = lane&15, K pair base = (lane>>4)*2  (v0=K, v1=K+1)
    //   B: lane holds N = lane&15, same K pair selection
    //   C/D: lane holds N = lane&15, VGPR v -> M = v + (lane>>4)*8
    const int mn   = lane & 15;
    const int koff = (lane >> 4) * 2;

    // ================= Phase 1: S = scale * Q K^T (causal) =================
    for (int j = wave; j < NKT; j += NWAVES) {
        if (j > qt) {
            // tile entirely above the causal diagonal
            #pragma unroll
            for (int v = 0; v < 8; ++v) {
                int m = v + (lane >> 4) * 8;
                Ssm[m * L_ + j * 16 + mn] = MASKF_;
            }
        } else {
            v8f c = {};
            const float* Krow = Kb + (size_t)(j * 16 + mn) * D_; // this lane's key row (N)
            #pragma unroll 8
            for (int k0 = 0; k0 < D_; k0 += 4) {
                v2f a = *(const v2f*)&Qsm[mn * D_ + k0 + koff];
                v2f b = *(const v2f*)&Krow[k0 + koff];
                c = __builtin_amdgcn_wmma_f32_16x16x4_f32(
                        false, a, false, b, (short)0, c, false, false);
            }
            #pragma unroll
            for (int v = 0; v < 8; ++v) {
                int m  = v + (lane >> 4) * 8;    // local query row
                int kg = j * 16 + mn;            // global key index
                int qg = qrow0 + m;              // global query index
                Ssm[m * L_ + j * 16 + mn] = (kg <= qg) ? c[v] * SCALE_ : MASKF_;
            }
        }
    }
    __syncthreads();

    // ================= Phase 2: row softmax, write attn =================
    {
        const int srow = tid >> 4;       // 0..15 (lanes 0-15 / 16-31 halves)
        const int scol = tid & 15;
        float* Srow = Ssm + srow * L_;
        const int kmax = (qt + 1) * 16;  // keys beyond this are fully masked

        float mx = -__builtin_inff();
        for (int c = scol; c < kmax; c += 16) mx = fmaxf(mx, Srow[c]);
        #pragma unroll
        for (int off = 1; off < 16; off <<= 1) mx = fmaxf(mx, __shfl_xor(mx, off));

        float sum = 0.0f;
        for (int c = scol; c < kmax; c += 16) {
            float e = __expf(Srow[c] - mx);  // masked entries underflow to 0
            Srow[c] = e;
            sum += e;
        }
        #pragma unroll
        for (int off = 1; off < 16; off <<= 1) sum += __shfl_xor(sum, off);
        const float inv = 1.0f / sum;

        float* arow = attn + (size_t)(bh * L_ + qrow0 + srow) * L_;
        for (int c = scol; c < kmax; c += 16) {
            float p = Srow[c] * inv;
            Srow[c] = p;        // keep normalized P in LDS for phase 3
            arow[c] = p;
        }
        for (int c = kmax + scol; c < L_; c += 16) arow[c] = 0.0f;
    }
    __syncthreads();

    // ================= Phase 3: context = P @ V =================
    {
        const int nW = wave * 16;              // this wave's d-column tile
        const int nchunks = (qt + 1) * 4;      // K chunks of 4 over causal extent
        v8f c = {};
        for (int j2 = 0; j2 < nchunks; ++j2) {
            int kb = j2 * 4 + koff;
            v2f a = *(const v2f*)&Ssm[mn * L_ + kb];   // P[m][kb], P[m][kb+1]
            v2f b;
            b.x = Vb[(size_t)(kb    ) * D_ + nW + mn]; // V[kb  ][n]
            b.y = Vb[(size_t)(kb + 1) * D_ + nW + mn]; // V[kb+1][n]
            c = __builtin_amdgcn_wmma_f32_16x16x4_f32(
                    false, a, false, b, (short)0, c, false, false);
        }
        #pragma unroll
        for (int v = 0; v < 8; ++v) {
            int m = v + (lane >> 4) * 8;
            ctx[(size_t)(bh * L_ + qrow0 + m) * D_ + nW + mn] = c[v];
        }
    }
}

extern "C" void kernel_launch(void* const* d_in, const int* in_sizes, int n_in,
                              void* d_out, int out_size, void* d_ws, size_t ws_size,
                              hipStream_t stream) {
    (void)in_sizes; (void)n_in; (void)out_size; (void)d_ws; (void)ws_size;
    const float* Q = (const float*)d_in[0];
    const float* K = (const float*)d_in[1];
    const float* V = (const float*)d_in[2];
    // d_in[3] is the boolean causal mask; implemented analytically in-kernel.

    float* ctx  = (float*)d_out;                                  // [B,H,L,D]
    float* attn = (float*)d_out + (size_t)B_ * H_ * L_ * D_;      // [B,H,L,L]

    const size_t shmem = (size_t)(16 * L_ + 16 * D_) * sizeof(float); // 136 KB
    hipFuncSetAttribute((const void*)sdpa_fused_f32wmma,
                        hipFuncAttributeMaxDynamicSharedMemorySize, (int)shmem);

    sdpa_fused_f32wmma<<<dim3(B_ * H_ * NKT), dim3(THREADS), shmem, stream>>>(
        Q, K, V, ctx, attn);
}